// MultiGraphSage_Dropout_50740743635552
// MI455X (gfx1250) — compile-verified
//
#include <hip/hip_runtime.h>
#include <math.h>

// Problem constants (match reference)
#define G 4
#define NN 50000
#define NE 800000
#define F 128

#define ROW_TILES ((NN + 15) / 16)          // 3125
#define WAVES_PER_WG 8
#define GEMM_BLOCKS ((ROW_TILES + WAVES_PER_WG - 1) / WAVES_PER_WG)  // 391

typedef float v2f __attribute__((ext_vector_type(2)));
typedef float v8f __attribute__((ext_vector_type(8)));

// ---------------------------------------------------------------------------
// Degree count: one thread per edge, count messages landing on dst.
// ---------------------------------------------------------------------------
__global__ void count_deg_kernel(const int* __restrict__ ei, float* __restrict__ cnt) {
    long long t = (long long)blockIdx.x * blockDim.x + threadIdx.x;
    if (t >= (long long)G * NE) return;
    int g = (int)(t / NE);
    int e = (int)(t % NE);
    int dst = ei[(size_t)g * 2 * NE + NE + e];
    atomicAdd(&cnt[(size_t)g * NN + dst], 1.0f);
}

// ---------------------------------------------------------------------------
// Edge scatter: 32 threads per edge, each moves a float4 (16B) chunk of the
// 128-float feature row feat[src] onto agg[dst] with fp32 atomics. Per-graph
// working set (~64 MB) is L2-resident on MI455X (192 MB L2).
// ---------------------------------------------------------------------------
__global__ void scatter_kernel(const float* __restrict__ feat,
                               const int* __restrict__ ei,
                               float* __restrict__ agg) {
    long long t = (long long)blockIdx.x * blockDim.x + threadIdx.x;
    if (t >= (long long)G * NE * 32) return;
    int chunk = (int)(t & 31);
    long long eg = t >> 5;
    int g = (int)(eg / NE);
    int e = (int)(eg % NE);
    int src = ei[(size_t)g * 2 * NE + e];
    int dst = ei[(size_t)g * 2 * NE + NE + e];
    const float4 v = *(const float4*)&feat[((size_t)g * NN + src) * F + chunk * 4];
    float* p = &agg[((size_t)g * NN + dst) * F + chunk * 4];
    atomicAdd(p + 0, v.x);
    atomicAdd(p + 1, v.y);
    atomicAdd(p + 2, v.z);
    atomicAdd(p + 3, v.w);
}

// ---------------------------------------------------------------------------
// Finalize mean in place: agg[node][f] /= max(cnt[node], 1)
// ---------------------------------------------------------------------------
__global__ void finalize_kernel(float* __restrict__ agg, const float* __restrict__ cnt) {
    long long t = (long long)blockIdx.x * blockDim.x + threadIdx.x;
    if (t >= (long long)G * NN * (F / 4)) return;
    size_t node = (size_t)t / (F / 4);
    float c = fmaxf(cnt[node], 1.0f);
    float4* p = (float4*)agg + t;
    float4 v = *p;
    v.x /= c; v.y /= c; v.z /= c; v.w /= c;
    *p = v;
}

// ---------------------------------------------------------------------------
// SAGE layer: out = tanh(agg @ Wl + self @ Wr + b) * mask, full fp32 via
// V_WMMA_F32_16X16X4_F32.
//
// 256 threads = 8 waves per workgroup. All waves share one 16-col weight
// slice (Wl,Wr : 128K x 16N = 2 x 8 KB) staged once in LDS; each wave then
// computes its own 16x16 output tile (8 consecutive row tiles per WG),
// feeding WMMA A from global (contiguous float2) and B from LDS.
//
// 32-bit WMMA layouts (ISA 7.12.2):
//   A (16x4):  lane l -> row = l&15, half = l>>4, element e -> K = 2*half + e
//   B (4x16):  lane l -> col = l&15,              element e -> K = 2*half + e
//   C/D:       lane l -> col = l&15, VGPR v -> row = v + 8*half
// LDS bank note: lanes 0-15 hit banks (16k+n)%64, lanes 16-31 the set +32:
// disjoint -> conflict-free ds reads.
// ---------------------------------------------------------------------------
__global__ __launch_bounds__(32 * WAVES_PER_WG)
void sage_wmma_kernel(const float* __restrict__ aggM, const float* __restrict__ self,
                      const float* __restrict__ Wl,   const float* __restrict__ Wr,
                      const float* __restrict__ bias, const float* __restrict__ mask,
                      float* __restrict__ out) {
    __shared__ float sWl[F * 16];   // [k][n], 8 KB
    __shared__ float sWr[F * 16];   // [k][n], 8 KB

    const int tid  = threadIdx.x;
    const int lane = tid & 31;
    const int wid  = tid >> 5;
    const int mrow = lane & 15;     // A row / B col / C col
    const int half = lane >> 4;
    const int g    = blockIdx.z;
    const int col0 = blockIdx.y * 16;

    // ---- Cooperative LDS fill: 2 float4 per thread per matrix ----
    const float* wlG = Wl + (size_t)g * F * F + col0;
    const float* wrG = Wr + (size_t)g * F * F + col0;
#pragma unroll
    for (int i = 0; i < 2; ++i) {
        const int idx = tid * 2 + i;        // 0..511 float4 slots
        const int k   = idx >> 2;           // 0..127
        const int n4  = (idx & 3) * 4;      // 0,4,8,12
        *(float4*)&sWl[k * 16 + n4] = *(const float4*)&wlG[(size_t)k * F + n4];
        *(float4*)&sWr[k * 16 + n4] = *(const float4*)&wrG[(size_t)k * F + n4];
    }
    __syncthreads();

    const size_t rowTile = (size_t)blockIdx.x * WAVES_PER_WG + wid;
    if (rowTile >= ROW_TILES) return;       // after barrier: safe divergence
    const size_t row0 = rowTile * 16;

    const float* aggRow  = aggM + ((size_t)g * NN + row0 + mrow) * F;
    const float* selfRow = self + ((size_t)g * NN + row0 + mrow) * F;

    // Bias is per output column -> same value in all 8 accumulator rows.
    const float bval = bias[(size_t)g * F + col0 + mrow];
    v8f c;
#pragma unroll
    for (int v = 0; v < 8; ++v) c[v] = bval;

#pragma unroll 8
    for (int kb = 0; kb < F; kb += 4) {
        const int k = kb + 2 * half;
        v2f aAgg  = *(const v2f*)(aggRow  + k);   // contiguous K,K+1
        v2f aSelf = *(const v2f*)(selfRow + k);
        v2f bWl, bWr;
        bWl[0] = sWl[(k)     * 16 + mrow];
        bWl[1] = sWl[(k + 1) * 16 + mrow];
        bWr[0] = sWr[(k)     * 16 + mrow];
        bWr[1] = sWr[(k + 1) * 16 + mrow];
        // (neg_a, A, neg_b, B, c_mod, C, reuse_a, reuse_b)
        c = __builtin_amdgcn_wmma_f32_16x16x4_f32(false, aAgg,  false, bWl,
                                                  (short)0, c, false, false);
        c = __builtin_amdgcn_wmma_f32_16x16x4_f32(false, aSelf, false, bWr,
                                                  (short)0, c, false, false);
    }

#pragma unroll
    for (int v = 0; v < 8; ++v) {
        const size_t m   = row0 + v + 8 * half;
        const size_t idx = ((size_t)g * NN + m) * F + col0 + mrow;
        out[idx] = tanhf(c[v]) * mask[idx];
    }
}

// ---------------------------------------------------------------------------
// Host-side launcher (graph-capture safe: only async memsets + kernels).
// ---------------------------------------------------------------------------
extern "C" void kernel_launch(void* const* d_in, const int* in_sizes, int n_in,
                              void* d_out, int out_size, void* d_ws, size_t ws_size,
                              hipStream_t stream) {
    const float* x   = (const float*)d_in[0];
    const int*   ei  = (const int*)  d_in[1];
    const float* Wl1 = (const float*)d_in[2];
    const float* Wr1 = (const float*)d_in[3];
    const float* b1  = (const float*)d_in[4];
    const float* Wl2 = (const float*)d_in[5];
    const float* Wr2 = (const float*)d_in[6];
    const float* b2  = (const float*)d_in[7];
    const float* m1  = (const float*)d_in[8];
    const float* m2  = (const float*)d_in[9];
    float* outp = (float*)d_out;

    // Workspace layout: agg (G*N*F f32) | cnt (G*N f32) | h (G*N*F f32)
    float* agg = (float*)d_ws;
    float* cnt = agg + (size_t)G * NN * F;
    float* h   = cnt + (size_t)G * NN;

    const size_t aggBytes = (size_t)G * NN * F * sizeof(float);
    const size_t cntBytes = (size_t)G * NN * sizeof(float);

    const long long nCnt  = (long long)G * NE;
    const long long nScat = (long long)G * NE * 32;
    const long long nFin  = (long long)G * NN * (F / 4);
    const int cntBlocks  = (int)((nCnt  + 255) / 256);
    const int scatBlocks = (int)((nScat + 255) / 256);
    const int finBlocks  = (int)((nFin  + 255) / 256);
    const dim3 gemmGrid(GEMM_BLOCKS, F / 16, G);
    const int  gemmBlock = 32 * WAVES_PER_WG;

    // ---- Layer 1 ----
    hipMemsetAsync(agg, 0, aggBytes, stream);
    hipMemsetAsync(cnt, 0, cntBytes, stream);
    count_deg_kernel<<<cntBlocks, 256, 0, stream>>>(ei, cnt);
    scatter_kernel<<<scatBlocks, 256, 0, stream>>>(x, ei, agg);
    finalize_kernel<<<finBlocks, 256, 0, stream>>>(agg, cnt);
    sage_wmma_kernel<<<gemmGrid, gemmBlock, 0, stream>>>(agg, x, Wl1, Wr1, b1, m1, h);

    // ---- Layer 2 (degrees unchanged; reuse cnt) ----
    hipMemsetAsync(agg, 0, aggBytes, stream);
    scatter_kernel<<<scatBlocks, 256, 0, stream>>>(h, ei, agg);
    finalize_kernel<<<finBlocks, 256, 0, stream>>>(agg, cnt);
    sage_wmma_kernel<<<gemmGrid, gemmBlock, 0, stream>>>(agg, h, Wl2, Wr2, b2, m2, outp);
}